// CustomKernel_1In1Out_Conv1D_34565896798381
// MI455X (gfx1250) — compile-verified
//
#include <hip/hip_runtime.h>
#include <hip/hip_bf16.h>

typedef __attribute__((ext_vector_type(2))) float v2f;
typedef __attribute__((ext_vector_type(4))) float v4f;
typedef __attribute__((ext_vector_type(8))) float v8f;

#define KSZ 16
#define LEN 8192
#define HID 64
#define OCH 8
#define OUT_LEN (LEN - KSZ + 1)      // 8177
#define TILES_PER_BATCH 512          // ceil(8177 / 16)
#define NBATCH 64
#define WAVES_PER_BLOCK 8
#define THREADS (WAVES_PER_BLOCK * 32)

// Branchless tanh on the transcendental unit: v_exp_f32 + v_rcp_f32.
// tanh(v) = 1 - 2/(exp2(v * 2*log2(e)) + 1)
// Saturates correctly: v >> 0 -> exp2 = +inf -> rcp = 0 -> 1; v << 0 -> exp2 = 0 -> -1.
__device__ __forceinline__ float fast_tanh(float v) {
    float t = __builtin_amdgcn_exp2f(v * 2.885390081777927f);   // 2/ln(2)
    float r = __builtin_amdgcn_rcpf(t + 1.0f);
    return 1.0f - 2.0f * r;
}

// One wave (32 lanes) processes a tile of 16 consecutive output positions of one
// batch row: im2col A(16x16) -> h(16x64) = tanh(A*W1 + b1) -> y(16x8) = h*W2 + b2.
// All matrix math goes through V_WMMA_F32_16X16X4_F32 (full f32 accumulation).
// EXEC is all-1s from entry through the last WMMA (no divergent code before them).
// Epilogue repacks the y tile through LDS so the wave emits one fully-coalesced
// global_store_b128 per lane (512B contiguous per wave) -- the output write
// stream (16.7 MB) is the roofline bottleneck at 23.3 TB/s.
__global__ __launch_bounds__(THREADS)
void conv1d_mlp_wmma(const float* __restrict__ x,
                     const float* __restrict__ W1,
                     const float* __restrict__ b1,
                     const float* __restrict__ W2,
                     const float* __restrict__ b2,
                     float* __restrict__ y)
{
    // Per-wave staging: h tile for the M<->N re-layout between GEMM1 (C/D layout,
    // lane-indexed by column) and GEMM2 (A layout, lane-indexed by row), plus a
    // small y-tile buffer for store coalescing.
    __shared__ float ldsH[WAVES_PER_BLOCK][16 * HID];   // 8 * 4KB = 32KB
    __shared__ float ldsY[WAVES_PER_BLOCK][16 * OCH];   // 8 * 512B = 4KB

    const int lane  = threadIdx.x & 31;
    const int wave  = threadIdx.x >> 5;
    const int tile  = blockIdx.x * WAVES_PER_BLOCK + wave;
    const int batch = tile >> 9;                 // / TILES_PER_BATCH
    const int t0    = (tile & (TILES_PER_BATCH - 1)) << 4;

    const int half = lane >> 4;                  // 0 -> lanes 0-15, 1 -> lanes 16-31
    const int lm   = lane & 15;                  // M (A frag) / N (B frag) / col (C,D)

    // ---------------- A fragments: im2col window of x -----------------------
    // Row m of the tile is x[batch, t0+m : t0+m+16]; the whole A tile lives in
    // 31 contiguous floats, so each lane's 16x4-f32 A fragment for K-step kk is
    // just a shifted 2-float load:  lane holds (M=lm, K=4kk+2*half .. +1).
    const float* xb = x + (long)batch * LEN;
    const int baseIdx = t0 + lm + 2 * half;
    v2f afrag[4];
    if (t0 + 31 < LEN) {                         // fast path: tile fully in-bounds
        #pragma unroll
        for (int kk = 0; kk < 4; ++kk) {
            const int idx = baseIdx + 4 * kk;
            afrag[kk].x = xb[idx];
            afrag[kk].y = xb[idx + 1];
        }
    } else {                                     // last tile of each batch: clamp
        #pragma unroll
        for (int kk = 0; kk < 4; ++kk) {
            const int idx = baseIdx + 4 * kk;
            const int i0 = (idx     < LEN) ? idx     : LEN - 1;
            const int i1 = (idx + 1 < LEN) ? idx + 1 : LEN - 1;
            afrag[kk].x = xb[i0];
            afrag[kk].y = xb[i1];
        }
    }

    // ---------------- B fragments of W1 (16x64), 4 N-tiles x 4 K-steps ------
    // B 4x16 f32 layout (ISA 7.12.2, symmetric to A): lane holds
    // (K = 4kk+2*half .. +1, N = 16n+lm).
    v2f b1frag[4][4];
    #pragma unroll
    for (int n = 0; n < 4; ++n) {
        #pragma unroll
        for (int kk = 0; kk < 4; ++kk) {
            const int row = 4 * kk + 2 * half;
            const int col = 16 * n + lm;
            b1frag[n][kk].x = W1[(row    ) * HID + col];
            b1frag[n][kk].y = W1[(row + 1) * HID + col];
        }
    }

    // ---------------- GEMM1 (K=16) + bias + tanh, spill h to LDS ------------
    float* hbuf = &ldsH[wave][0];
    #pragma unroll
    for (int n = 0; n < 4; ++n) {
        v8f acc = {};
        #pragma unroll
        for (int kk = 0; kk < 4; ++kk) {
            acc = __builtin_amdgcn_wmma_f32_16x16x4_f32(
                      false, afrag[kk], false, b1frag[n][kk],
                      (short)0, acc, false, false);
        }
        const float bias = b1[16 * n + lm];
        #pragma unroll
        for (int i = 0; i < 8; ++i) {
            const int M = i + 8 * half;          // C/D layout: vgpr i, half -> row
            hbuf[M * HID + 16 * n + lm] = fast_tanh(acc[i] + bias);
        }
    }

    // ---------------- B fragments of W2 (64x8 widened to 64x16, cols 8..15 = cols 0..7) ---------
    // y columns 8..15 are never stored, so lanes 8..15 can load don't-care
    // weights (duplicate of column lm&7) -- keeps this branch-free.
    v2f b2frag[16];
    #pragma unroll
    for (int kk = 0; kk < 16; ++kk) {
        const int row = 4 * kk + 2 * half;
        const int col = lm & (OCH - 1);
        b2frag[kk].x = W2[(row    ) * OCH + col];
        b2frag[kk].y = W2[(row + 1) * OCH + col];
    }

    // ---------------- GEMM2 (K=64): h as A fragments from LDS ---------------
    // DS ops are in-order within a wave, so the re-layout needs no barrier.
    v8f accY = {};
    #pragma unroll
    for (int kk = 0; kk < 16; ++kk) {
        const int kcol = 4 * kk + 2 * half;
        v2f a2;
        a2.x = hbuf[lm * HID + kcol];
        a2.y = hbuf[lm * HID + kcol + 1];
        accY = __builtin_amdgcn_wmma_f32_16x16x4_f32(
                   false, a2, false, b2frag[kk], (short)0, accY, false, false);
    }

    // ---------------- bias, repack y tile in LDS, coalesced store -----------
    // Spill the 16x8 valid tile (C/D layout -> row-major) to LDS, then each of
    // the 32 lanes stores 16 contiguous bytes: one global_store_b128 covering
    // the wave's whole 512B output span with zero gaps.
    const float bias2 = b2[lm & (OCH - 1)];      // branch-free bias fetch
    float* ybuf = &ldsY[wave][0];
    if (lm < OCH) {
        #pragma unroll
        for (int i = 0; i < 8; ++i) {
            const int M = i + 8 * half;
            ybuf[M * OCH + lm] = accY[i] + bias2;
        }
    }
    // Same wave's DS pipeline is in-order: loads below see the stores above.
    const int row = lane >> 1;                   // output row this lane covers
    v4f out4;
    out4.x = ybuf[lane * 4 + 0];
    out4.y = ybuf[lane * 4 + 1];
    out4.z = ybuf[lane * 4 + 2];
    out4.w = ybuf[lane * 4 + 3];
    if (t0 + row < OUT_LEN) {
        const long base = (long)batch * OUT_LEN + t0;
        v4f* dst = (v4f*)(y + (base + row) * OCH + (lane & 1) * 4);
        *dst = out4;
    }
}

extern "C" void kernel_launch(void* const* d_in, const int* in_sizes, int n_in,
                              void* d_out, int out_size, void* d_ws, size_t ws_size,
                              hipStream_t stream) {
    (void)in_sizes; (void)n_in; (void)d_ws; (void)ws_size; (void)out_size;
    const float* x  = (const float*)d_in[0];
    const float* W1 = (const float*)d_in[1];
    const float* b1 = (const float*)d_in[2];
    const float* W2 = (const float*)d_in[3];
    const float* b2 = (const float*)d_in[4];
    float* y = (float*)d_out;

    const int total_tiles = NBATCH * TILES_PER_BATCH;       // 32768
    const int blocks = total_tiles / WAVES_PER_BLOCK;       // 4096
    conv1d_mlp_wmma<<<blocks, THREADS, 0, stream>>>(x, W1, b1, W2, b2, y);
}